// SharpnessLoss_58136677319222
// MI455X (gfx1250) — compile-verified
//
#include <hip/hip_runtime.h>

// Problem constants (fixed by the reference harness)
#define B_ 128
#define L_ 512
#define S_ 512
#define C_ 64
#define EPS_ 1e-6f

typedef float v2f __attribute__((ext_vector_type(2)));
typedef float v8f __attribute__((ext_vector_type(8)));

// ---- ordered float <-> uint keys for atomicMax on floats -------------------
__device__ __forceinline__ unsigned fkey(float f) {
  unsigned u = __float_as_uint(f);
  return (u & 0x80000000u) ? ~u : (u | 0x80000000u);
}
__device__ __forceinline__ float fval(unsigned kk) {
  unsigned u = (kk & 0x80000000u) ? (kk & 0x7fffffffu) : ~kk;
  return __uint_as_float(u);
}

// ---- fragment loaders ------------------------------------------------------
// A frag (16x4 f32): lane l holds A[l%16][2*(l/16)], A[l%16][2*(l/16)+1]
// B frag (4x16 f32): lane l holds B[2*(l/16)][l%16], B[2*(l/16)+1][l%16]
//                    = k[n=l%16][c+2h], k[n][c+2h+1]  (same pattern as A)
__device__ __forceinline__ void load_bfrag(v2f bf[16], const float* kb, int n0,
                                           int row16, int half) {
  const float* kp = kb + (n0 + row16) * C_ + 2 * half;
#pragma unroll
  for (int kk = 0; kk < 16; ++kk) bf[kk] = *(const v2f*)(kp + kk * 4);
}

// ---- 16 chained V_WMMA_F32_16X16X4_F32 over K=64, two independent chains ---
__device__ __forceinline__ v8f wmma_chain(const v2f areg[16], const v2f bf[16]) {
  v8f acc0 = {0.f, 0.f, 0.f, 0.f, 0.f, 0.f, 0.f, 0.f};
  v8f acc1 = {0.f, 0.f, 0.f, 0.f, 0.f, 0.f, 0.f, 0.f};
#pragma unroll
  for (int kk = 0; kk < 16; kk += 2) {
    acc0 = __builtin_amdgcn_wmma_f32_16x16x4_f32(
        false, areg[kk], false, bf[kk], (short)0, acc0, false, false);
    acc1 = __builtin_amdgcn_wmma_f32_16x16x4_f32(
        false, areg[kk + 1], false, bf[kk + 1], (short)0, acc1, false, false);
  }
  return acc0 + acc1;
}

// ---- kernel 1: zero accumulators (must run every launch: graph replay) -----
__global__ void k_init(float* colSum, float* colSq, unsigned* colMaxK,
                       float* lossAcc) {
  int i = blockIdx.x * blockDim.x + threadIdx.x;
  if (i < B_ * S_) {
    colSum[i] = 0.f;
    colSq[i] = 0.f;
    colMaxK[i] = 0u;  // 0 < fkey(x) for any finite x
  }
  if (i < B_ * 4) lossAcc[i] = 0.f;
}

// ---- kernel 2: GEMM + row/col stats ----------------------------------------
// grid = B*32 (batch, row-block of 16 rows), block = 128 threads (4 waves)
__global__ void __launch_bounds__(128)
k_stats(const float* __restrict__ q, const float* __restrict__ k,
        float* __restrict__ rowMax, float* __restrict__ rowSum,
        float* __restrict__ rowSq, float* __restrict__ colSum,
        float* __restrict__ colSq, unsigned* __restrict__ colMaxK) {
  const int b = blockIdx.x >> 5;
  const int rb = blockIdx.x & 31;
  const int lane = threadIdx.x & 31;
  const int wave = threadIdx.x >> 5;
  const int row16 = lane & 15;
  const int half = lane >> 4;
  const int m0 = rb * 16;

  const float* qb = q + (size_t)b * L_ * C_;
  const float* kb = k + (size_t)b * S_ * C_;

  v2f areg[16];
#pragma unroll
  for (int kk = 0; kk < 16; ++kk)
    areg[kk] = *(const v2f*)(qb + (m0 + row16) * C_ + kk * 4 + 2 * half);

  float rMax[8], rSum[8], rSq[8];
#pragma unroll
  for (int v = 0; v < 8; ++v) { rMax[v] = -3.4e38f; rSum[v] = 0.f; rSq[v] = 0.f; }

  // software-pipelined tile loop: load tile i+1's B frags while tile i's
  // WMMA chain runs (avoids the load->wait0->wmma serialization)
  v2f buf0[16], buf1[16];
  load_bfrag(buf0, kb, wave * 16, row16, half);
#pragma unroll
  for (int i = 0; i < 8; ++i) {
    const int t = wave + i * 4;
    const int n0 = t * 16;
    v8f acc;
    if (i & 1) {
      if (i < 7) load_bfrag(buf0, kb, (t + 4) * 16, row16, half);
      acc = wmma_chain(areg, buf1);
    } else {
      if (i < 7) load_bfrag(buf1, kb, (t + 4) * 16, row16, half);
      acc = wmma_chain(areg, buf0);
    }
    // C/D layout: VGPR v, lane l: m = v + 8*(l/16), n = l%16
    float cMax = -3.4e38f, cSum = 0.f, cSq = 0.f;
#pragma unroll
    for (int v = 0; v < 8; ++v) {
      float s = 0.5f + 0.5f * acc[v];
      rMax[v] = fmaxf(rMax[v], s);
      rSum[v] += s;
      rSq[v] += s * s;
      cMax = fmaxf(cMax, s);
      cSum += s;
      cSq += s * s;
    }
    // column partial: combine the two lane-halves (same n, different m-sets)
    cMax = fmaxf(cMax, __shfl_xor(cMax, 16));
    cSum += __shfl_xor(cSum, 16);
    cSq += __shfl_xor(cSq, 16);
    if (lane < 16) {
      int idx = b * S_ + n0 + lane;
      atomicAdd(&colSum[idx], cSum);
      atomicAdd(&colSq[idx], cSq);
      atomicMax(&colMaxK[idx], fkey(cMax));
    }
  }

  // row partials: reduce across the 16 lanes inside each half (same m)
#pragma unroll
  for (int v = 0; v < 8; ++v) {
#pragma unroll
    for (int off = 1; off < 16; off <<= 1) {
      rMax[v] = fmaxf(rMax[v], __shfl_xor(rMax[v], off));
      rSum[v] += __shfl_xor(rSum[v], off);
      rSq[v] += __shfl_xor(rSq[v], off);
    }
  }
  __shared__ float sMax[4][16], sSum[4][16], sSq[4][16];
  if (row16 == 0) {  // lanes 0 (rows v) and 16 (rows v+8) of each wave
#pragma unroll
    for (int v = 0; v < 8; ++v) {
      int r = v + 8 * half;
      sMax[wave][r] = rMax[v];
      sSum[wave][r] = rSum[v];
      sSq[wave][r] = rSq[v];
    }
  }
  __syncthreads();
  if (threadIdx.x < 16) {  // rows are exclusive to this block: plain store
    int r = threadIdx.x;
    float m = sMax[0][r], s = sSum[0][r], q2 = sSq[0][r];
#pragma unroll
    for (int w = 1; w < 4; ++w) {
      m = fmaxf(m, sMax[w][r]);
      s += sSum[w][r];
      q2 += sSq[w][r];
    }
    int idx = b * L_ + m0 + r;
    rowMax[idx] = m;
    rowSum[idx] = s;
    rowSq[idx] = q2;
  }
}

// ---- kernel 3: scores + softmax (sharp1 -> d_out, sharp2 -> ws) ------------
__global__ void __launch_bounds__(512)
k_softmax(const float* __restrict__ colSum, const float* __restrict__ colSq,
          const unsigned* __restrict__ colMaxK, const float* __restrict__ rowSum,
          const float* __restrict__ rowSq, const float* __restrict__ rowMax,
          float* __restrict__ sharp1, float* __restrict__ sharp2) {
  __shared__ float red[512];
  const int tid = threadIdx.x;
  const int bb = blockIdx.x;
  float score;
  float* outp;
  int idx;
  if (bb < B_) {  // column direction -> sharp1
    idx = bb * S_ + tid;
    float sum = colSum[idx], sq = colSq[idx];
    float mean = sum * (1.0f / 512.0f);
    float var = (sq - sum * sum * (1.0f / 512.0f)) * (1.0f / 511.0f);  // ddof=1
    score = (fval(colMaxK[idx]) - mean) / sqrtf(fmaxf(var, 1e-30f));
    outp = sharp1;
  } else {  // row direction -> sharp2
    idx = (bb - B_) * L_ + tid;
    float sum = rowSum[idx], sq = rowSq[idx];
    float mean = sum * (1.0f / 512.0f);
    float var = (sq - sum * sum * (1.0f / 512.0f)) * (1.0f / 511.0f);
    score = (rowMax[idx] - mean) / sqrtf(fmaxf(var, 1e-30f));
    outp = sharp2;
  }
  red[tid] = score;
  __syncthreads();
  for (int o = 256; o > 0; o >>= 1) {
    if (tid < o) red[tid] = fmaxf(red[tid], red[tid + o]);
    __syncthreads();
  }
  float m = red[0];
  __syncthreads();
  float e = __expf(score - m);
  red[tid] = e;
  __syncthreads();
  for (int o = 256; o > 0; o >>= 1) {
    if (tid < o) red[tid] += red[tid + o];
    __syncthreads();
  }
  outp[idx] = e / red[0];
}

// ---- kernel 4: recompute GEMM + masked loss sums ---------------------------
__global__ void __launch_bounds__(128)
k_loss(const float* __restrict__ q, const float* __restrict__ k,
       const long long* __restrict__ labels, const float* __restrict__ sharp1,
       const float* __restrict__ sharp2, float* __restrict__ lossAcc) {
  const int b = blockIdx.x >> 5;
  const int rb = blockIdx.x & 31;
  const int lane = threadIdx.x & 31;
  const int wave = threadIdx.x >> 5;
  const int row16 = lane & 15;
  const int half = lane >> 4;
  const int m0 = rb * 16;

  const float* qb = q + (size_t)b * L_ * C_;
  const float* kb = k + (size_t)b * S_ * C_;

  v2f areg[16];
#pragma unroll
  for (int kk = 0; kk < 16; ++kk)
    areg[kk] = *(const v2f*)(qb + (m0 + row16) * C_ + kk * 4 + 2 * half);

  float w2[8];
#pragma unroll
  for (int v = 0; v < 8; ++v) w2[v] = sharp2[b * L_ + m0 + v + 8 * half];

  // low dword of each little-endian int64 label (values are 0/1)
  const int* labp =
      (const int*)(labels + ((size_t)b * L_ + m0 + 8 * half) * S_ + row16);

  float a1 = 0.f, a2 = 0.f, a3 = 0.f, aN = 0.f;

  v2f buf0[16], buf1[16];
  load_bfrag(buf0, kb, wave * 16, row16, half);
#pragma unroll
  for (int i = 0; i < 8; ++i) {
    const int t = wave + i * 4;
    const int n0 = t * 16;
    // issue HBM-bound label loads first so they hide behind the WMMA chain
    int labv[8];
#pragma unroll
    for (int v = 0; v < 8; ++v) labv[v] = labp[(v * S_ + n0) * 2];
    float w1 = sharp1[b * S_ + n0 + row16];  // n fixed per lane within tile
    v8f acc;
    if (i & 1) {
      if (i < 7) load_bfrag(buf0, kb, (t + 4) * 16, row16, half);
      acc = wmma_chain(areg, buf1);
    } else {
      if (i < 7) load_bfrag(buf1, kb, (t + 4) * 16, row16, half);
      acc = wmma_chain(areg, buf0);
    }
#pragma unroll
    for (int v = 0; v < 8; ++v) {
      int lab = labv[v];
      float sim = 0.5f + 0.5f * acc[v];
      float simc = fminf(fmaxf(sim, EPS_), 1.0f - EPS_);
      bool isNeg = (lab == 0);
      float arg = isNeg ? (1.0f - simc) : simc;  // one log per element
      float nl = -__logf(arg);
      if (lab == 1) {
        a1 += nl * w1;
        a2 += nl * w2[v];
      } else if (isNeg) {
        a3 += nl;
        aN += 1.0f;
      }
    }
  }
#pragma unroll
  for (int off = 1; off < 32; off <<= 1) {
    a1 += __shfl_xor(a1, off);
    a2 += __shfl_xor(a2, off);
    a3 += __shfl_xor(a3, off);
    aN += __shfl_xor(aN, off);
  }
  if (lane == 0) {
    atomicAdd(&lossAcc[b * 4 + 0], a1);
    atomicAdd(&lossAcc[b * 4 + 1], a2);
    atomicAdd(&lossAcc[b * 4 + 2], a3);
    atomicAdd(&lossAcc[b * 4 + 3], aN);
  }
}

// ---- kernel 5: combine + mean over batches ---------------------------------
__global__ void __launch_bounds__(128) k_final(const float* __restrict__ lossAcc,
                                               float* __restrict__ out) {
  __shared__ float red[128];
  const int b = threadIdx.x;
  float a1 = lossAcc[b * 4 + 0], a2 = lossAcc[b * 4 + 1];
  float a3 = lossAcc[b * 4 + 2], aN = lossAcc[b * 4 + 3];
  red[b] = 0.5f * (a1 + a2) + a3 / fmaxf(aN, 1.0f);
  __syncthreads();
  for (int o = 64; o > 0; o >>= 1) {
    if (b < o) red[b] += red[b + o];
    __syncthreads();
  }
  if (b == 0) out[B_ * S_] = red[0] * (1.0f / 128.0f);
}

extern "C" void kernel_launch(void* const* d_in, const int* in_sizes, int n_in,
                              void* d_out, int out_size, void* d_ws,
                              size_t ws_size, hipStream_t stream) {
  const float* q = (const float*)d_in[0];
  const float* k = (const float*)d_in[1];
  const long long* labels = (const long long*)d_in[2];
  float* out = (float*)d_out;  // [B*S] sharp1, then 1 scalar mean loss

  char* ws = (char*)d_ws;
  const size_t n1 = (size_t)B_ * S_;  // == B_*L_
  float* colSum = (float*)ws;        ws += n1 * 4;
  float* colSq = (float*)ws;         ws += n1 * 4;
  unsigned* colMaxK = (unsigned*)ws; ws += n1 * 4;
  float* rowSum = (float*)ws;        ws += n1 * 4;
  float* rowSq = (float*)ws;         ws += n1 * 4;
  float* rowMax = (float*)ws;        ws += n1 * 4;
  float* sharp2 = (float*)ws;        ws += n1 * 4;
  float* lossAcc = (float*)ws;       ws += (size_t)B_ * 4 * 4;
  (void)ws_size; (void)in_sizes; (void)n_in; (void)out_size;

  hipLaunchKernelGGL(k_init, dim3((B_ * S_ + 255) / 256), dim3(256), 0, stream,
                     colSum, colSq, colMaxK, lossAcc);
  hipLaunchKernelGGL(k_stats, dim3(B_ * 32), dim3(128), 0, stream, q, k, rowMax,
                     rowSum, rowSq, colSum, colSq, colMaxK);
  hipLaunchKernelGGL(k_softmax, dim3(2 * B_), dim3(512), 0, stream, colSum,
                     colSq, colMaxK, rowSum, rowSq, rowMax, out, sharp2);
  hipLaunchKernelGGL(k_loss, dim3(B_ * 32), dim3(128), 0, stream, q, k, labels,
                     out, sharp2, lossAcc);
  hipLaunchKernelGGL(k_final, dim3(1), dim3(128), 0, stream, lossAcc, out);
}